// RelativePosition_25082609009118
// MI455X (gfx1250) — compile-verified
//
#include <hip/hip_runtime.h>

// Problem constants (match the reference)
#define B_   2
#define H_   8
#define L_   32
#define CH   64          // chunk size
#define D_   64
#define N_   (L_ * CH)   // 2048
#define MAXREL 32

typedef float v2f __attribute__((ext_vector_type(2)));
typedef float v4f __attribute__((ext_vector_type(4)));
typedef float v8f __attribute__((ext_vector_type(8)));

// LDS strides (floats), padded for bank-conflict-free access + 16B alignment
#define XSTR 68   // x chunk rows
#define ESTR 68   // embedding rows
#define SSTR 36   // score rows

__global__ __launch_bounds__(256)
void RelativePosition_25082609009118_kernel(const float* __restrict__ x,
                                            const long long* __restrict__ pos,
                                            const float* __restrict__ table,
                                            float* __restrict__ out)
{
    __shared__ float lx[CH * XSTR];   // x chunk   [64][68]
    __shared__ float le[L_ * ESTR];   // emb rows  [32][68]
    __shared__ float ls[CH * SSTR];   // scores    [64][36]

    const int bid  = blockIdx.x;          // b*H*L + h*L + q
    const int b    = bid / (H_ * L_);
    const int h    = (bid / L_) % H_;
    const int q    = bid % L_;
    const int t    = threadIdx.x;
    const int lane = t & 31;
    const int wave = t >> 5;              // 0..7

    // ---- Stage x chunk [64][64] into LDS (b128, coalesced) ----
    const float* xsrc = x + (((size_t)(b * H_ + h)) * N_ + (size_t)q * CH) * D_;
#pragma unroll
    for (int i = 0; i < 4; ++i) {
        int e   = t + 256 * i;            // 0..1023 float4 slots
        int row = e >> 4;                 // 0..63
        int c4  = (e & 15) << 2;          // 0..60 step 4
        v4f v = *(const v4f*)(xsrc + row * D_ + c4);
        *(v4f*)(&lx[row * XSTR + c4]) = v;
    }

    // ---- Gather emb rows: le[k][d] = table[clip(pos[b,q,k])+32][d] ----
    const long long* prow = pos + ((size_t)b * L_ + q) * L_;
#pragma unroll
    for (int i = 0; i < 2; ++i) {
        int e   = t + 256 * i;            // 0..511
        int row = e >> 4;                 // 0..31
        int c4  = (e & 15) << 2;
        long long p = prow[row];
        long long pc = p < -MAXREL ? -MAXREL : (p > MAXREL ? MAXREL : p);
        int idx = (int)pc + MAXREL;       // 0..64
        v4f v = *(const v4f*)(table + (size_t)idx * D_ + c4);
        *(v4f*)(&le[row * ESTR + c4]) = v;
    }
    __syncthreads();

    // ---- WMMA: scores[n][k] = sum_d A[n][d] * emb[k][d]  (64x32, f32) ----
    // One 16x16 tile per wave: 4 M-tiles x 2 N-tiles.
    const int m0   = (wave >> 1) << 4;    // 0,16,32,48  (rows n)
    const int n0   = (wave & 1) << 4;     // 0,16        (cols k)
    const int half = lane >> 4;           // 0/1
    const int l16  = lane & 15;

    v8f acc = {};
    const float* arow = &lx[(m0 + l16) * XSTR + 2 * half];  // A: M=lane%16, K=2*half+v
    const float* brow = &le[(n0 + l16) * ESTR + 2 * half];  // B: N=lane%16, K=2*half+v
#pragma unroll
    for (int kk = 0; kk < 16; ++kk) {     // K = 64 in steps of 4
        v2f a, bv;
        a.x  = arow[4 * kk + 0];
        a.y  = arow[4 * kk + 1];
        bv.x = brow[4 * kk + 0];
        bv.y = brow[4 * kk + 1];
        // (neg_a, A, neg_b, B, c_mod, C, reuse_a, reuse_b)
        acc = __builtin_amdgcn_wmma_f32_16x16x4_f32(false, a, false, bv,
                                                    (short)0, acc, false, false);
    }

    // D layout: VGPR r -> row m0 + r + 8*half, col n0 + l16
#pragma unroll
    for (int r = 0; r < 8; ++r) {
        ls[(m0 + r + 8 * half) * SSTR + (n0 + l16)] = acc[r];
    }
    __syncthreads();

    // ---- Broadcast store: 64 rows x 2048 cols, each score repeated 64x ----
    // 256 MB total stream-out > 192 MB L2: non-temporal stores keep the
    // write stream from thrashing L2. Consecutive lanes -> consecutive
    // float4s -> fully coalesced global_store_b128.
    v4f* ob = (v4f*)(out + (((size_t)(b * H_ + h)) * N_ + (size_t)q * CH) * (size_t)N_);
#pragma unroll
    for (int i = 0; i < 128; ++i) {
        int e   = t + 256 * i;            // 0..32767 float4 slots in 64x2048 block
        int row = e >> 9;                 // 0..63
        int c4  = e & 511;                // float4 within row
        float s = ls[row * SSTR + (c4 >> 4)];
        v4f v = {s, s, s, s};
        __builtin_nontemporal_store(v, ob + (size_t)e);
    }
}

extern "C" void kernel_launch(void* const* d_in, const int* in_sizes, int n_in,
                              void* d_out, int out_size, void* d_ws, size_t ws_size,
                              hipStream_t stream) {
    const float*     x     = (const float*)d_in[0];       // [B,H,N,D] fp32
    const long long* pos   = (const long long*)d_in[1];   // [B,L,L] int64
    const float*     table = (const float*)d_in[2];       // [65,D] fp32
    // d_in[3] = chunk_size scalar (compile-time constant here)
    float* out = (float*)d_out;                           // [B,H,N,N] fp32

    dim3 grid(B_ * H_ * L_);   // 512 workgroups, one per (b,h,q)
    dim3 block(256);           // 8 waves of 32
    RelativePosition_25082609009118_kernel<<<grid, block, 0, stream>>>(x, pos, table, out);
}